// SAGE_39427799777330
// MI455X (gfx1250) — compile-verified
//
#include <hip/hip_runtime.h>

typedef __attribute__((ext_vector_type(2))) float v2f;
typedef __attribute__((ext_vector_type(8))) float v8f;

#define K_FEATS 512

// ---------------------------------------------------------------------------
// Scatter-add aggregation: 128 threads (4 waves) cover one edge's 512 features
// as float4 -> fully coalesced 2KB row read + contiguous f32 atomic adds that
// resolve in L2 (agg buffers fit in the 192MB L2).
// ---------------------------------------------------------------------------
__global__ __launch_bounds__(256) void sage_scatter(
    const float* __restrict__ Hsrc,
    const int* __restrict__ src,
    const int* __restrict__ dst,
    float* __restrict__ Agg,
    float* __restrict__ Deg,
    int nEdges)
{
    long long tid = (long long)blockIdx.x * blockDim.x + threadIdx.x;
    int e = (int)(tid >> 7);                  // 128 threads per edge
    if (e >= nEdges) return;
    int f = ((int)tid & 127) << 2;            // feature offset (float4 granule)
    int s = src[e];
    int d = dst[e];
    const float4 v = *(const float4*)(Hsrc + (size_t)s * K_FEATS + f);
    float* q = Agg + (size_t)d * K_FEATS + f;
    atomicAdd(q + 0, v.x);
    atomicAdd(q + 1, v.y);
    atomicAdd(q + 2, v.z);
    atomicAdd(q + 3, v.w);
    if (f == 0) atomicAdd(Deg + d, 1.0f);     // one lane per edge counts degree
}

// ---------------------------------------------------------------------------
// Fused SAGE GEMM:  Out = relu?( Hself @ Wself + diag(1/deg) Agg @ Wneigh + b )
// One wave per 16x16 C tile, dual accumulators, V_WMMA_F32_16X16X4_F32.
//   A fragment (16x4):  lane l -> row l&15, K = k + 2*(l>>4) + j  (b64 load)
//   B fragment (4x16):  VGPR j -> K = k + 2*(l>>4) + j, N = lane&15
//   C fragment (16x16): VGPR v -> row v + 8*(l>>4), col lane&15
// N is a template parameter: all strides are immediates; for N=47 the
// out-of-range columns clamp their load address (their C columns are garbage
// but never stored), so EXEC stays all-1s with zero predication overhead.
// ---------------------------------------------------------------------------
template<int RELU, int N>
__global__ __launch_bounds__(128) void sage_gemm_wmma(
    const float* __restrict__ Hself,
    const float* __restrict__ Agg,
    const float* __restrict__ Deg,
    const float* __restrict__ Wself,
    const float* __restrict__ Wneigh,
    const float* __restrict__ Bias,
    float* __restrict__ Out)
{
    constexpr int NT = (N + 15) / 16;         // N tiles (compile-time)

    const int wave = blockIdx.x * (blockDim.x >> 5) + (threadIdx.x >> 5);
    const int mt = wave / NT;
    const int nt = wave % NT;

    const int lane = threadIdx.x & 31;
    const int g    = lane >> 4;               // lane group (0: lanes 0-15, 1: 16-31)
    const int ln   = lane & 15;
    const int row  = mt * 16 + ln;            // A-fragment row owned by this lane
    const int col  = nt * 16 + ln;            // C column owned by this lane
    const int colc = (col < N) ? col : (N - 1);  // clamped for loads (N=47 only)

    v8f cs = {0.f, 0.f, 0.f, 0.f, 0.f, 0.f, 0.f, 0.f};  // self accumulator
    v8f cn = {0.f, 0.f, 0.f, 0.f, 0.f, 0.f, 0.f, 0.f};  // neighbor accumulator

    // Running pointers: immediate-offset addressing after unrolling.
    const float* ap = Hself  + (size_t)row * K_FEATS + 2 * g;
    const float* gp = Agg    + (size_t)row * K_FEATS + 2 * g;
    const float* bs = Wself  + (size_t)(2 * g) * N + colc;
    const float* bn = Wneigh + (size_t)(2 * g) * N + colc;

    #pragma unroll 8
    for (int k = 0; k < K_FEATS; k += 4) {
        v2f a0 = *(const v2f*)ap;  ap += 4;
        v2f a1 = *(const v2f*)gp;  gp += 4;
        v2f b0 = { bs[0], bs[N] }; bs += 4 * N;
        v2f b1 = { bn[0], bn[N] }; bn += 4 * N;
        cs = __builtin_amdgcn_wmma_f32_16x16x4_f32(false, a0, false, b0,
                                                   (short)0, cs, false, false);
        cn = __builtin_amdgcn_wmma_f32_16x16x4_f32(false, a1, false, b1,
                                                   (short)0, cn, false, false);
    }

    // Row-wise 1/deg applied once at the end (commutes with the GEMM).
    const float bias = Bias[colc];
    #pragma unroll
    for (int v = 0; v < 8; ++v) {
        int r = mt * 16 + v + 8 * g;          // C row for this VGPR slot
        float rdeg = 1.0f / fmaxf(Deg[r], 1.0f);
        float val = cs[v] + cn[v] * rdeg + bias;
        if (RELU) val = fmaxf(val, 0.f);
        if (col < N) Out[(size_t)r * N + col] = val;
    }
}

// ---------------------------------------------------------------------------
// Host-side orchestration (graph-capture safe: only async memset + launches)
// ---------------------------------------------------------------------------
extern "C" void kernel_launch(void* const* d_in, const int* in_sizes, int n_in,
                              void* d_out, int out_size, void* d_ws, size_t ws_size,
                              hipStream_t stream) {
    (void)in_sizes; (void)n_in; (void)out_size; (void)ws_size;

    const float* x       = (const float*)d_in[0];
    const int*   src0    = (const int*)d_in[1];
    const int*   dst0    = (const int*)d_in[2];
    const int*   src1    = (const int*)d_in[3];
    const int*   dst1    = (const int*)d_in[4];
    const int*   src2    = (const int*)d_in[5];
    const int*   dst2    = (const int*)d_in[6];
    const float* Wself0  = (const float*)d_in[7];
    const float* Wneigh0 = (const float*)d_in[8];
    const float* b0      = (const float*)d_in[9];
    const float* Wself1  = (const float*)d_in[10];
    const float* Wneigh1 = (const float*)d_in[11];
    const float* b1      = (const float*)d_in[12];
    const float* Wself2  = (const float*)d_in[13];
    const float* Wneigh2 = (const float*)d_in[14];
    const float* b2      = (const float*)d_in[15];
    float* out = (float*)d_out;

    const int ND0 = 51200, ND1 = 5120, ND2 = 512;        // NUM_DST (compile-time)
    const int NE0 = ND0 * 10, NE1 = ND1 * 10, NE2 = ND2 * 10;

    // Workspace carve-up (agg_i immediately followed by deg_i -> single memset)
    float* ws = (float*)d_ws;
    size_t o = 0;
    float* agg0 = ws + o; o += (size_t)ND0 * K_FEATS;
    float* deg0 = ws + o; o += ND0;
    float* agg1 = ws + o; o += (size_t)ND1 * K_FEATS;
    float* deg1 = ws + o; o += ND1;
    float* agg2 = ws + o; o += (size_t)ND2 * K_FEATS;
    float* deg2 = ws + o; o += ND2;
    float* h1   = ws + o; o += (size_t)ND0 * K_FEATS;    // layer-0 output
    float* h2   = ws + o; o += (size_t)ND1 * K_FEATS;    // layer-1 output

    // ---------------- Layer 0: 51200x512 out, ReLU ----------------
    hipMemsetAsync(agg0, 0, ((size_t)ND0 * K_FEATS + ND0) * sizeof(float), stream);
    sage_scatter<<<(NE0 * 128) / 256, 256, 0, stream>>>(x, src0, dst0, agg0, deg0, NE0);
    sage_gemm_wmma<1, 512><<<(ND0 / 16) * (512 / 16) / 4, 128, 0, stream>>>(
        x, agg0, deg0, Wself0, Wneigh0, b0, h1);

    // ---------------- Layer 1: 5120x512 out, ReLU ----------------
    hipMemsetAsync(agg1, 0, ((size_t)ND1 * K_FEATS + ND1) * sizeof(float), stream);
    sage_scatter<<<(NE1 * 128) / 256, 256, 0, stream>>>(h1, src1, dst1, agg1, deg1, NE1);
    sage_gemm_wmma<1, 512><<<(ND1 / 16) * (512 / 16) / 4, 128, 0, stream>>>(
        h1, agg1, deg1, Wself1, Wneigh1, b1, h2);

    // ---------------- Layer 2: 512x47 out, no ReLU ----------------
    hipMemsetAsync(agg2, 0, ((size_t)ND2 * K_FEATS + ND2) * sizeof(float), stream);
    sage_scatter<<<(NE2 * 128) / 256, 256, 0, stream>>>(h2, src2, dst2, agg2, deg2, NE2);
    sage_gemm_wmma<0, 47><<<(ND2 / 16) * 3 / 4, 128, 0, stream>>>(
        h2, agg2, deg2, Wself2, Wneigh2, b2, out);
}